// RotationScaleAdapter_62294205662124
// MI455X (gfx1250) — compile-verified
//
#include <hip/hip_runtime.h>
#include <math.h>
#include <stdint.h>

// Problem constants from the reference: B=4, S=8192, D=1024.
#define DDIM 1024
#define WAVE 32
#define EPL  32               // elements per lane = DDIM / WAVE
#define ROWS_PER_BLOCK 8      // 8 waves per 256-thread block, one row per wave

// ---------------------------------------------------------------------------
// Kernel 1: precompute {exp(log_scale), cos(theta), sin(theta), 0} once into
// d_ws as float4, ALREADY TRANSPOSED: element j lives at (j%32)*32 + (j/32).
// The main kernel DMA-stages this tile into LDS (async-to-LDS engine) and
// each lane's per-step read is a single conflict-free ds_load_b128.
// ---------------------------------------------------------------------------
__global__ __launch_bounds__(256)
void rot_params_kernel(const float* __restrict__ log_scale,
                       const float* __restrict__ theta,
                       float4* __restrict__ ws4) {
    int i = blockIdx.x * 256 + threadIdx.x;
    if (i < DDIM) {
        int tj = ((i & 31) << 5) | (i >> 5);
        ws4[tj] = make_float4(expf(log_scale[i]), cosf(theta[i]), sinf(theta[i]), 0.0f);
    }
}

// ---------------------------------------------------------------------------
// Kernel 2: per-row Givens cascade via a parallel affine-map scan, wave32/row.
//
// With v = x*escale, carry_0 = v[0], for i = 0..D-2:
//   out_i       = cos_i*carry_i - sin_i*v[i+1]
//   carry_{i+1} = sin_i*carry_i + cos_i*v[i+1]
// Wrap: y[D-1] = c_last*carry_{D-1} - s_last*out_0
//       y[0]   = s_last*carry_{D-1} + c_last*out_0
// Output = y - x.
//
// Lane L owns i in [32L, 32L+32). One fused pass expresses each output as an
// affine function of the lane's incoming carry (out = P[k]*cin + Q[k]) while
// composing the lane's step map (A,B). A 5-step wave shuffle-scan resolves
// cin; outputs are then one independent FMA + SUB per element.
// ---------------------------------------------------------------------------
__global__ __launch_bounds__(256)
void rot_cascade_kernel(const float* __restrict__ x,
                        const float4* __restrict__ params,
                        float* __restrict__ out,
                        int rows) {
    __shared__ float4 pcsT[DDIM];   // 16 KiB: {esc, cos, sin, 0}, transposed

    const int t = threadIdx.x;

    // ---- async DMA staging: global -> LDS, no VGPR round trip (ASYNCcnt) ----
#pragma unroll
    for (int m = 0; m < DDIM / 256; ++m) {
        int j = t + m * 256;
        unsigned lds_off = (unsigned)(uintptr_t)&pcsT[j];   // low 32b = LDS addr
        unsigned goff    = (unsigned)(j * (int)sizeof(float4));
        asm volatile("global_load_async_to_lds_b128 %0, %1, %2"
                     :: "v"(lds_off), "v"(goff), "s"(params)
                     : "memory");
    }

    const int  lane = t & 31;
    const int  wave = t >> 5;
    const long row  = (long)blockIdx.x * ROWS_PER_BLOCK + wave;

    const float* __restrict__ xrow = x   + row * DDIM + lane * EPL;
    float*       __restrict__ orow = out + row * DDIM + lane * EPL;

    // gfx1250 global_prefetch_b8: start pulling the row while the DMA runs.
    if (row < rows) __builtin_prefetch(xrow, 0, 0);

    // my async transfers done -> then make all waves' LDS writes visible
    asm volatile("s_wait_asynccnt 0x0" ::: "memory");
    __syncthreads();
    if (row >= rows) return;        // whole wave exits together (after barrier)

    // ---- load row: 8 x b128 per lane, fully aligned ----
    float xv[EPL];
#pragma unroll
    for (int q = 0; q < 8; ++q) {
        float4 f = ((const float4*)xrow)[q];
        xv[4*q+0] = f.x; xv[4*q+1] = f.y; xv[4*q+2] = f.z; xv[4*q+3] = f.w;
    }

    // ---- single fused pass: build P/Q per element + compose lane map ----
    float P[EPL], Q[EPL];
    float A = 1.0f, B = 0.0f;

    float4 f0   = pcsT[lane];               // k = 0 -> {esc,c,s} of index 32L
    float v0loc = xv[0] * f0.x;             // v[32L]
    float cP = f0.y, sP = f0.z;             // pipelined (c,s) of current step

    // v[32(L+1)] = lane (L+1)'s v0loc (lane 31: unused garbage)
    float vnext = __shfl_down(v0loc, 1, WAVE);

#pragma unroll
    for (int k = 1; k < EPL; ++k) {
        float4 g  = pcsT[(k << 5) | lane];  // one ds_load_b128 per element
        float  vn = xv[k] * g.x;            // v[32L+k]
        // step (k-1): out = cP*(A*cin + B) - sP*vn
        P[k-1] = cP * A;
        Q[k-1] = fmaf(cP, B, -(sP * vn));
        B = fmaf(sP, B, cP * vn);
        A = sP * A;
        cP = g.y; sP = g.z;
    }
    // step 31 (i = 32L+31); i = 1023 does not exist -> lane 31 skips compose
    P[EPL-1] = cP * A;
    Q[EPL-1] = fmaf(cP, B, -(sP * vnext));  // lane 31: garbage, overwritten by wrap
    if (lane < WAVE - 1) {
        B = fmaf(sP, B, cP * vnext);
        A = sP * A;
    }

    // ---- inclusive Hillis-Steele scan of affine maps across the wave ----
#pragma unroll
    for (int off = 1; off < WAVE; off <<= 1) {
        float Ap = __shfl_up(A, off, WAVE);
        float Bp = __shfl_up(B, off, WAVE);
        if (lane >= off) {
            B = fmaf(A, Bp, B);             // current ∘ previous
            A = A * Ap;
        }
    }

    float v0 = __shfl(v0loc, 0, WAVE);      // carry_0 = v[0]
    // lane 31 inclusive map covers steps 0..1022 -> carry_{1023} directly
    float carry_end = fmaf(A, v0, B);
    float carryF    = __shfl(carry_end, WAVE - 1, WAVE);

    // exclusive scan -> carry entering this lane
    float Aex = __shfl_up(A, 1, WAVE);
    float Bex = __shfl_up(B, 1, WAVE);
    float cin = (lane == 0) ? v0 : fmaf(Aex, v0, Bex);

    // ---- outputs: one independent FMA + SUB per element ----
    float x0s  = xv[0];
    float x31s = xv[EPL-1];
    float out0_raw = fmaf(P[0], cin, Q[0]); // raw out_0 (meaningful on lane 0)
#pragma unroll
    for (int k = 0; k < EPL; ++k)
        xv[k] = fmaf(P[k], cin, Q[k]) - xv[k];

    // ---- wrap step: positions 0 and D-1 ----
    float out0 = __shfl(out0_raw, 0, WAVE);
    float4 last = pcsT[DDIM - 1];           // j=1023 -> tj=1023 (broadcast read)
    float cl = last.y, sl = last.z;
    float new_first = fmaf(sl, carryF,  cl * out0);
    float new_last  = fmaf(cl, carryF, -(sl * out0));
    if (lane == 0)        xv[0]     = new_first - x0s;
    if (lane == WAVE - 1) xv[EPL-1] = new_last  - x31s;

    // ---- store: 8 x b128 per lane ----
#pragma unroll
    for (int q = 0; q < 8; ++q) {
        float4 f;
        f.x = xv[4*q+0]; f.y = xv[4*q+1]; f.z = xv[4*q+2]; f.w = xv[4*q+3];
        ((float4*)orow)[q] = f;
    }
}

extern "C" void kernel_launch(void* const* d_in, const int* in_sizes, int n_in,
                              void* d_out, int out_size, void* d_ws, size_t ws_size,
                              hipStream_t stream) {
    const float* x         = (const float*)d_in[0];
    const float* log_scale = (const float*)d_in[1];
    const float* theta     = (const float*)d_in[2];
    float*  out = (float*)d_out;
    float4* ws4 = (float4*)d_ws;            // 16 KiB used

    rot_params_kernel<<<(DDIM + 255) / 256, 256, 0, stream>>>(log_scale, theta, ws4);

    const int rows   = in_sizes[0] / DDIM;                            // 32768
    const int blocks = (rows + ROWS_PER_BLOCK - 1) / ROWS_PER_BLOCK;  // 4096
    rot_cascade_kernel<<<blocks, 256, 0, stream>>>(x, ws4, out, rows);
}